// sce_function_69630009803211
// MI455X (gfx1250) — compile-verified
//
#include <hip/hip_runtime.h>
#include <stdint.h>

#define NBINS 15
#define BLOCK 256
#define WAVES (BLOCK / 32)
#define PX_PER_THREAD 4
#define DEPTH 4                 // async pipeline depth (staging buffers)
#define CSUM_SCALE 4194304.0f   // 2^22 fixed-point scale for conf sums

// Packed per-element contribution: count [63:50], acc [49:36], csum_fp [35:0].
// Per-(wave,bin) capacity: 16383 elements; csum < 16383 * 2^22 < 2^36. One
// ds_add_u64 per element-channel, exactly order-independent => deterministic.

#if __has_builtin(__builtin_amdgcn_s_wait_asynccnt)
#define WAIT_ASYNC(n) __builtin_amdgcn_s_wait_asynccnt(n)
#else
#define WAIT_ASYNC(n) asm volatile("s_wait_asynccnt %0" ::"n"(n) : "memory")
#endif

// CDNA5 GLOBAL_LOAD_ASYNC_TO_LDS_B128 (GV addressing), tracked by ASYNCcnt.
__device__ __forceinline__ void async_copy_b128(uint32_t lds_addr, const void* gaddr) {
  asm volatile("global_load_async_to_lds_b128 %0, %1, off"
               :: "v"(lds_addr), "v"(gaddr)
               : "memory");
}

// searchsorted(linspace(0,1,16), c, side='left') - 1  ==  ceil(15*c) - 1
__device__ __forceinline__ void accum(unsigned long long* hc, float c, int acc) {
  int b = (int)ceilf(c * 15.0f) - 1;
  if ((unsigned)b < (unsigned)NBINS) {
    unsigned long long v = (1ull << 50)
                         | ((unsigned long long)(unsigned)acc << 36)
                         | (unsigned long long)(c * CSUM_SCALE);
    atomicAdd(&hc[b], v);
  }
}

__global__ void __launch_bounds__(BLOCK) hist_kernel(
    const float* __restrict__ probs, const int* __restrict__ labels,
    size_t npix, int iters, float* __restrict__ partials) {
  // Async staging: [buf][wave][stream][lane][4 dwords] = 48 KB
  __shared__ alignas(16) float stage[DEPTH][WAVES][3][32][4];
  // Wave-private packed histograms: [wave][ch][bin] = 2 KB
  __shared__ unsigned long long hist[WAVES][2][16];

  const int tid  = threadIdx.x;
  const int wv   = tid >> 5;
  const int lane = tid & 31;

  ((unsigned long long*)hist)[tid] = 0ull;  // 256 u64 with 256 threads
  __syncthreads();

  const float* conf1 = probs + npix;
  const size_t chunk = (size_t)gridDim.x * (BLOCK * PX_PER_THREAD);

  // Per-lane LDS byte addresses (low 32 bits of a generic LDS pointer).
  uint32_t lds_c0[DEPTH], lds_c1[DEPTH], lds_lb[DEPTH];
  for (int b = 0; b < DEPTH; ++b) {
    lds_c0[b] = (uint32_t)(uintptr_t)&stage[b][wv][0][lane][0];
    lds_c1[b] = (uint32_t)(uintptr_t)&stage[b][wv][1][lane][0];
    lds_lb[b] = (uint32_t)(uintptr_t)&stage[b][wv][2][lane][0];
  }

  // Rolling global pointers advanced by `chunk` per issued stage.
  const size_t base0 = (size_t)blockIdx.x * (BLOCK * PX_PER_THREAD)
                     + (size_t)tid * PX_PER_THREAD;
  const float* pc0 = probs  + base0;
  const float* pc1 = conf1  + base0;
  const int*   plb = labels + base0;

  auto issue = [&](int buf) {
    async_copy_b128(lds_c0[buf], pc0);
    async_copy_b128(lds_c1[buf], pc1);
    async_copy_b128(lds_lb[buf], plb);
    pc0 += chunk; pc1 += chunk; plb += chunk;
  };

  // Prologue: fill up to DEPTH-1 stages.
  for (int j = 0; j < DEPTH - 1 && j < iters; ++j) issue(j);

  for (int i = 0; i < iters; ++i) {
    const int cur = i & (DEPTH - 1);
    const int newer = iters - 1 - i;      // stages in flight beyond current
    if (i + DEPTH - 1 < iters) {
      issue((i + DEPTH - 1) & (DEPTH - 1));
      WAIT_ASYNC(3 * (DEPTH - 1));        // 9: 3 newer stages may be pending
    } else if (newer >= 2) {
      WAIT_ASYNC(6);
    } else if (newer == 1) {
      WAIT_ASYNC(3);
    } else {
      WAIT_ASYNC(0);
    }

    const float4 c0 = *(const float4*)&stage[cur][wv][0][lane][0];
    const float4 c1 = *(const float4*)&stage[cur][wv][1][lane][0];
    const int4   lb = *(const int4  *)&stage[cur][wv][2][lane][0];

    accum(hist[wv][0], c0.x, lb.x == 0);
    accum(hist[wv][0], c0.y, lb.y == 0);
    accum(hist[wv][0], c0.z, lb.z == 0);
    accum(hist[wv][0], c0.w, lb.w == 0);

    accum(hist[wv][1], c1.x, lb.x == 1);
    accum(hist[wv][1], c1.y, lb.y == 1);
    accum(hist[wv][1], c1.z, lb.z == 1);
    accum(hist[wv][1], c1.w, lb.w == 1);
  }

  // Tail (inactive for the reference shape: exact division holds).
  const size_t done = (size_t)iters * chunk;
  for (size_t p = done + (size_t)blockIdx.x * BLOCK + tid; p < npix;
       p += (size_t)gridDim.x * BLOCK) {
    float a = probs[p], b = conf1[p];
    int l = labels[p];
    accum(hist[wv][0], a, l == 0);
    accum(hist[wv][1], b, l == 1);
  }

  __syncthreads();

  // Cross-wave merge -> this block's private 96-float slice of partials.
  // Layout: [ch][stat][bin16], stat: 0=count, 1=csum, 2=asum.
  if (tid < 96) {
    const int ch = tid / 48, r = tid % 48, stat = r / 16, bin = r % 16;
    unsigned long long cnt = 0, acc = 0, cs = 0;
    for (int w = 0; w < WAVES; ++w) {
      const unsigned long long s = hist[w][ch][bin];
      cnt += s >> 50;
      acc += (s >> 36) & 0x3FFFull;
      cs  += s & 0xFFFFFFFFFull;  // low 36 bits
    }
    float v;
    if (stat == 0)      v = (float)cnt;
    else if (stat == 2) v = (float)acc;
    else                v = (float)((double)cs * (1.0 / 4194304.0));
    partials[(size_t)blockIdx.x * 96 + tid] = v;
  }
}

__global__ void finalize_kernel(const float* __restrict__ partials, int nblocks,
                                float* __restrict__ out) {
  const int t = threadIdx.x;
  if (t < 90) {
    const int ch = t / 45, r = t % 45, stat = r / 15, bin = r % 15;
    float s = 0.f;
    for (int g = 0; g < nblocks; ++g)
      s += partials[(size_t)g * 96 + ch * 48 + stat * 16 + bin];
    out[t] = s;
  }
}

extern "C" void kernel_launch(void* const* d_in, const int* in_sizes, int n_in,
                              void* d_out, int out_size, void* d_ws, size_t ws_size,
                              hipStream_t stream) {
  const float* probs  = (const float*)d_in[0];
  const int*   labels = (const int*)d_in[1];
  const size_t npix   = (size_t)in_sizes[1];  // 4096*4096 pixels

  float* partials = (float*)d_ws;
  int G = 1024;  // 1024 WGs * 256 thr * 4 px = 1,048,576 px/stage -> 16 stages
  while ((size_t)G * 96 * sizeof(float) > ws_size && G > 1) G >>= 1;
  const int iters = (int)(npix / ((size_t)G * BLOCK * PX_PER_THREAD));

  hist_kernel<<<G, BLOCK, 0, stream>>>(probs, labels, npix, iters, partials);
  finalize_kernel<<<1, 128, 0, stream>>>(partials, G, (float*)d_out);
}